// HyperGraph_23699629539750
// MI455X (gfx1250) — compile-verified
//
#include <hip/hip_runtime.h>

#define FEAT 128
#define N_NODES_C 50000
#define N_HEDGES_C 10000

typedef float v2f __attribute__((ext_vector_type(2)));
typedef float v8f __attribute__((ext_vector_type(8)));

// Guaranteed hardware f32 atomic add (no CAS loop). Pointers are always
// device-global here, so the global_ form is safe. gfx1250 syntax:
//   global_atomic_add_f32 v[addr:addr+1], vdata, off
__device__ __forceinline__ void atomic_add_f32(float* p, float v) {
  asm volatile("global_atomic_add_f32 %0, %1, off"
               :
               : "v"(p), "v"(v)
               : "memory");
}

__global__ void k_zero(float* __restrict__ p, int n) {
  int i = blockIdx.x * blockDim.x + threadIdx.x;
  if (i < n) p[i] = 0.0f;
}

__global__ void k_degrees(const int* __restrict__ src, const int* __restrict__ dst,
                          float* __restrict__ deg_n, float* __restrict__ deg_e, int nnz) {
  int e = blockIdx.x * blockDim.x + threadIdx.x;
  if (e < nnz) {
    atomic_add_f32(&deg_n[src[e]], 1.0f);
    atomic_add_f32(&deg_e[dst[e]], 1.0f);
  }
}

__global__ void k_invert(float* __restrict__ d, int n) {
  int i = blockIdx.x * blockDim.x + threadIdx.x;
  if (i < n) {
    float v = d[i];
    d[i] = (v > 0.0f) ? (1.0f / v) : 0.0f;
  }
}

// C[M x 128] = A[M x 128] @ W[128 x 128], fp32 via V_WMMA_F32_16X16X4_F32.
// One wave (32 threads) per 16x16 output tile; K fully unrolled (32 WMMAs).
// A 16x4 f32 layout: lanes 0-15 hold {K=k0,k0+1} (VGPR0,1), lanes 16-31 hold
// {K=k0+2,k0+3}  -> contiguous float2 per lane. B 4x16 mirrored across lanes.
__global__ void k_gemm_wmma(const float* __restrict__ A, const float* __restrict__ W,
                            float* __restrict__ C) {
  const int lane = threadIdx.x;       // 0..31
  const int tm   = blockIdx.x;        // M/16 tiles
  const int tn   = blockIdx.y;        // 128/16 = 8 tiles
  const int l15  = lane & 15;
  const int hi   = lane >> 4;         // 0 or 1
  const int m    = tm * 16 + l15;
  const int n    = tn * 16 + l15;

  v8f acc = {};
#pragma unroll
  for (int k0 = 0; k0 < FEAT; k0 += 4) {
    const int k = k0 + 2 * hi;
    v2f a = *(const v2f*)(A + (size_t)m * FEAT + k);
    v2f b;
    b.x = W[(size_t)k * FEAT + n];
    b.y = W[(size_t)(k + 1) * FEAT + n];
    // (neg_a, A, neg_b, B, c_mod, C, reuse_a, reuse_b)
    acc = __builtin_amdgcn_wmma_f32_16x16x4_f32(false, a, false, b, (short)0, acc,
                                                false, false);
  }
  // C/D layout: VGPR r -> M = r + 8*hi, N = l15
  const int mrow = tm * 16 + 8 * hi;
#pragma unroll
  for (int r = 0; r < 8; ++r)
    C[(size_t)(mrow + r) * FEAT + tn * 16 + l15] = acc[r];
}

// he[dst[e]] += h[src[e]] ; one wave per edge, 4 floats per lane (float4 gather)
__global__ void k_scatter_n2e(const float* __restrict__ h, const int* __restrict__ src,
                              const int* __restrict__ dst, float* __restrict__ he,
                              int nnz) {
  int idx  = blockIdx.x * blockDim.x + threadIdx.x;
  int e    = idx >> 5;
  int lane = idx & 31;
  if (e >= nnz) return;
  int s = src[e], d = dst[e];
  const float4 v = *(const float4*)(h + (size_t)s * FEAT + lane * 4);
  float* p = he + (size_t)d * FEAT + lane * 4;
  atomic_add_f32(p + 0, v.x);
  atomic_add_f32(p + 1, v.y);
  atomic_add_f32(p + 2, v.z);
  atomic_add_f32(p + 3, v.w);
}

// accum[src[e]] += Binv[dst[e]] * he[dst[e]]   (B^-1 folded into this pass)
__global__ void k_scatter_e2n(const float* __restrict__ he, const float* __restrict__ Binv,
                              const int* __restrict__ src, const int* __restrict__ dst,
                              float* __restrict__ accum, int nnz) {
  int idx  = blockIdx.x * blockDim.x + threadIdx.x;
  int e    = idx >> 5;
  int lane = idx & 31;
  if (e >= nnz) return;
  int s = src[e], d = dst[e];
  float bw = Binv[d];
  const float4 v = *(const float4*)(he + (size_t)d * FEAT + lane * 4);
  float* p = accum + (size_t)s * FEAT + lane * 4;
  atomic_add_f32(p + 0, v.x * bw);
  atomic_add_f32(p + 1, v.y * bw);
  atomic_add_f32(p + 2, v.z * bw);
  atomic_add_f32(p + 3, v.w * bw);
}

// out = Dinv[node] * accum + bias  (+ optional ReLU), float4 per thread
__global__ void k_finalize(const float* __restrict__ accum, const float* __restrict__ Dinv,
                           const float* __restrict__ bias, float* __restrict__ out,
                           int n_nodes, int do_relu) {
  int idx  = blockIdx.x * blockDim.x + threadIdx.x;
  int node = idx >> 5;
  int lane = idx & 31;
  if (node >= n_nodes) return;
  float di = Dinv[node];
  const float4 v = *(const float4*)(accum + (size_t)node * FEAT + lane * 4);
  const float4 b = *(const float4*)(bias + lane * 4);
  float4 r;
  r.x = di * v.x + b.x;
  r.y = di * v.y + b.y;
  r.z = di * v.z + b.z;
  r.w = di * v.w + b.w;
  if (do_relu) {
    r.x = fmaxf(r.x, 0.0f);
    r.y = fmaxf(r.y, 0.0f);
    r.z = fmaxf(r.z, 0.0f);
    r.w = fmaxf(r.w, 0.0f);
  }
  *(float4*)(out + (size_t)node * FEAT + lane * 4) = r;
}

extern "C" void kernel_launch(void* const* d_in, const int* in_sizes, int n_in,
                              void* d_out, int out_size, void* d_ws, size_t ws_size,
                              hipStream_t stream) {
  const float* x  = (const float*)d_in[0];
  const int* ei   = (const int*)d_in[1];
  const float* W1 = (const float*)d_in[2];
  const float* b1 = (const float*)d_in[3];
  const float* W2 = (const float*)d_in[4];
  const float* b2 = (const float*)d_in[5];

  const int n_nodes  = in_sizes[0] / FEAT;   // 50000
  const int nnz      = in_sizes[1] / 2;      // 600000
  const int n_hedges = N_HEDGES_C;           // 10000 (reference constant)
  const int* src = ei;                       // edge_index[0]
  const int* dst = ei + nnz;                 // edge_index[1]

  // Workspace: bufA/bufB [N*D], he [H*D], deg_n [N], deg_e [H]
  float* bufA  = (float*)d_ws;
  float* bufB  = bufA + (size_t)n_nodes * FEAT;
  float* he    = bufB + (size_t)n_nodes * FEAT;
  float* deg_n = he + (size_t)n_hedges * FEAT;
  float* deg_e = deg_n + n_nodes;

  const int BT = 256;
  const int nd      = n_nodes * FEAT;
  const int hd      = n_hedges * FEAT;
  const int edgeThr = nnz * 32;
  dim3 gemmGrid(n_nodes / 16, FEAT / 16);

  // ---- degrees (recomputed every call: deterministic, ws is not preserved) ----
  k_zero<<<(n_nodes + n_hedges + BT - 1) / BT, BT, 0, stream>>>(deg_n, n_nodes + n_hedges);
  k_degrees<<<(nnz + BT - 1) / BT, BT, 0, stream>>>(src, dst, deg_n, deg_e, nnz);
  k_invert<<<(n_nodes + n_hedges + BT - 1) / BT, BT, 0, stream>>>(deg_n, n_nodes + n_hedges);

  // ---- layer 1 ----
  k_gemm_wmma<<<gemmGrid, 32, 0, stream>>>(x, W1, bufA);
  k_zero<<<(hd + BT - 1) / BT, BT, 0, stream>>>(he, hd);
  k_scatter_n2e<<<(edgeThr + BT - 1) / BT, BT, 0, stream>>>(bufA, src, dst, he, nnz);
  k_zero<<<(nd + BT - 1) / BT, BT, 0, stream>>>(bufB, nd);
  k_scatter_e2n<<<(edgeThr + BT - 1) / BT, BT, 0, stream>>>(he, deg_e, src, dst, bufB, nnz);
  k_finalize<<<(n_nodes * 32 + BT - 1) / BT, BT, 0, stream>>>(bufB, deg_n, b1, bufA,
                                                             n_nodes, /*relu=*/1);

  // ---- layer 2 ----
  k_gemm_wmma<<<gemmGrid, 32, 0, stream>>>(bufA, W2, bufB);
  k_zero<<<(hd + BT - 1) / BT, BT, 0, stream>>>(he, hd);
  k_scatter_n2e<<<(edgeThr + BT - 1) / BT, BT, 0, stream>>>(bufB, src, dst, he, nnz);
  k_zero<<<(nd + BT - 1) / BT, BT, 0, stream>>>(bufA, nd);
  k_scatter_e2n<<<(edgeThr + BT - 1) / BT, BT, 0, stream>>>(he, deg_e, src, dst, bufA, nnz);
  k_finalize<<<(n_nodes * 32 + BT - 1) / BT, BT, 0, stream>>>(bufA, deg_n, b2,
                                                             (float*)d_out, n_nodes,
                                                             /*relu=*/0);
}